// Switcher_17480516895517
// MI455X (gfx1250) — compile-verified
//
#include <hip/hip_runtime.h>
#include <hip/hip_bf16.h>
#include <math.h>

// Problem sizes (fixed by the reference)
#define S_DIM   512
#define B_DIM   32
#define D_DIM   1024
#define DOUT    1024

// Tiling for the main fused GEMM
#define TM      128    // s rows per workgroup (two 16-row tiles per wave, stacked)
#define TN      32     // f cols per workgroup
#define KC      64     // k chunk staged in LDS
#define LPITCH  68     // padded LDS row pitch (floats), even & mult-of-4

typedef __attribute__((ext_vector_type(2))) float v2f;
typedef __attribute__((ext_vector_type(8))) float v8f;

// ---------------------------------------------------------------------------
// Kernel 1: gating vectors
//   v1[b,d] = dot(x[0,b,:], w1_w[d,:]) + w1_b[d]   (b<B, d<D)
//   v2[b,f] = dot(x[0,b,:], w2_w[f,:]) + w2_b[f]   (b<B, f<DOUT)
// One thread per output element; everything is L2-resident (x0=128KB, w=8MB).
// ---------------------------------------------------------------------------
__global__ __launch_bounds__(256) void gate_kernel(
    const float* __restrict__ x,
    const float* __restrict__ w1_w, const float* __restrict__ w1_b,
    const float* __restrict__ w2_w, const float* __restrict__ w2_b,
    float* __restrict__ v1, float* __restrict__ v2)
{
  int idx = blockIdx.x * blockDim.x + threadIdx.x;   // 0 .. B*(D+DOUT)-1
  int half = idx / (B_DIM * D_DIM);                  // 0 -> v1, 1 -> v2
  int e    = idx - half * (B_DIM * D_DIM);
  int b    = e / D_DIM;
  int r    = e - b * D_DIM;                          // d or f

  const float4* xr = (const float4*)(x + (size_t)b * D_DIM);   // x[0,b,:]
  const float4* wr = (const float4*)((half ? w2_w : w1_w) + (size_t)r * D_DIM);

  float acc = 0.0f;
#pragma unroll 4
  for (int j = 0; j < D_DIM / 4; ++j) {
    float4 a = xr[j];
    float4 w = wr[j];
    acc = fmaf(a.x, w.x, acc);
    acc = fmaf(a.y, w.y, acc);
    acc = fmaf(a.z, w.z, acc);
    acc = fmaf(a.w, w.w, acc);
  }
  acc += half ? w2_b[r] : w1_b[r];
  if (half) v2[b * DOUT + r] = acc;
  else      v1[b * D_DIM + r] = acc;
}

// ---------------------------------------------------------------------------
// Kernel 2: fused gated GEMM with f32 WMMA
//   out[s,b,f] = sum_d x[s,b,d] * ( model_w[f,d] * sigmoid(v1[b,d]*v2[b,f]) )
//              + model_b[f]
// Workgroup: 256 threads = 8 waves, computes a TM(128) x TN(32) tile for one b.
// Waves arranged 4(s) x 2(f); each wave owns TWO 16x16 C tiles stacked in M
// (s offsets sw and sw+64) that share the same B (mask) fragment.
// The mask tile is computed on the fly into LDS once per k-chunk (fast
// v_rcp_f32 sigmoid) and reused by all 8 M-subtiles; never hits HBM.
// ---------------------------------------------------------------------------
__global__ __launch_bounds__(256) void gated_gemm_kernel(
    const float* __restrict__ x,
    const float* __restrict__ v1, const float* __restrict__ v2,
    const float* __restrict__ model_w, const float* __restrict__ model_b,
    float* __restrict__ out)
{
  __shared__ float Alds[TM][LPITCH];   // x tile   [s][k]
  __shared__ float Blds[TN][LPITCH];   // mask tile[f][k]

  const int b  = blockIdx.z;
  const int s0 = blockIdx.y * TM;
  const int f0 = blockIdx.x * TN;

  const int t    = threadIdx.x;
  const int lane = t & 31;
  const int wave = t >> 5;
  const int sw   = (wave & 3) * 16;    // this wave's s offset (tile0); tile1 at +64
  const int fw   = (wave >> 2) * 16;   // this wave's f offset

  const int lr   = lane & 15;          // row/col selector within fragment
  const int kh   = (lane >> 4) * 2;    // K pair base: lanes 0-15 -> K{0,1}, 16-31 -> K{2,3}

  v8f c0 = {0.f, 0.f, 0.f, 0.f, 0.f, 0.f, 0.f, 0.f};
  v8f c1 = {0.f, 0.f, 0.f, 0.f, 0.f, 0.f, 0.f, 0.f};

  for (int k0 = 0; k0 < D_DIM; k0 += KC) {
    __syncthreads();   // previous chunk's LDS reads done before overwrite

    // ---- stage A tile: x[s0..s0+127, b, k0..k0+63]  (2048 float4, 8/thread)
#pragma unroll
    for (int i = 0; i < 8; ++i) {
      int idx = t + i * 256;
      int r   = idx >> 4;          // 0..127
      int c4  = (idx & 15) * 4;    // 0..60
      float4 val = *(const float4*)(x + (size_t)(s0 + r) * (B_DIM * D_DIM)
                                      + (size_t)b * D_DIM + k0 + c4);
      *(float4*)&Alds[r][c4] = val;
    }

    // ---- stage B tile: mask[k0..k0+63, f0..f0+31], stored [f][k]
    //      mask = model_w[f, k] * sigmoid(v1[b,k] * v2[b,f])
    //      sigmoid via fast reciprocal (v_rcp_f32, <=1 ulp)
#pragma unroll
    for (int i = 0; i < 2; ++i) {
      int idx = t + i * 256;
      int f   = idx >> 4;          // 0..31
      int c4  = (idx & 15) * 4;    // 0..60
      float4 w  = *(const float4*)(model_w + (size_t)(f0 + f) * D_DIM + k0 + c4);
      float4 g1 = *(const float4*)(v1 + (size_t)b * D_DIM + k0 + c4);
      float  g2 = v2[(size_t)b * DOUT + f0 + f];
      float4 m;
      m.x = w.x * __builtin_amdgcn_rcpf(1.0f + __expf(-(g1.x * g2)));
      m.y = w.y * __builtin_amdgcn_rcpf(1.0f + __expf(-(g1.y * g2)));
      m.z = w.z * __builtin_amdgcn_rcpf(1.0f + __expf(-(g1.z * g2)));
      m.w = w.w * __builtin_amdgcn_rcpf(1.0f + __expf(-(g1.w * g2)));
      *(float4*)&Blds[f][c4] = m;
    }

    __syncthreads();

    // ---- 32 x V_WMMA_F32_16X16X4_F32 over this k chunk (2 C tiles / wave,
    //      sharing the B fragment: 3 ds b64-pair loads per 2 WMMAs)
#pragma unroll
    for (int k = 0; k < KC; k += 4) {
      v2f a0 = *(const v2f*)&Alds[sw + lr][k + kh];        // tile0 rows
      v2f a1 = *(const v2f*)&Alds[sw + 64 + lr][k + kh];   // tile1 rows
      v2f bb = *(const v2f*)&Blds[fw + lr][k + kh];        // shared B frag
      c0 = __builtin_amdgcn_wmma_f32_16x16x4_f32(
              false, a0, false, bb, (short)0, c0, false, false);
      c1 = __builtin_amdgcn_wmma_f32_16x16x4_f32(
              false, a1, false, bb, (short)0, c1, false, false);
    }
  }

  // ---- epilogue: C layout (VGPR r: lanes0-15 M=r, lanes16-31 M=8+r), add bias
  const int srow = s0 + sw + (lane >> 4) * 8;
  const int fcol = f0 + fw + lr;
  const float bias = model_b[fcol];
#pragma unroll
  for (int r = 0; r < 8; ++r) {
    out[(size_t)(srow + r) * (B_DIM * DOUT) + (size_t)b * DOUT + fcol] = c0[r] + bias;
  }
#pragma unroll
  for (int r = 0; r < 8; ++r) {
    out[(size_t)(srow + 64 + r) * (B_DIM * DOUT) + (size_t)b * DOUT + fcol] = c1[r] + bias;
  }
}

// ---------------------------------------------------------------------------
extern "C" void kernel_launch(void* const* d_in, const int* in_sizes, int n_in,
                              void* d_out, int out_size, void* d_ws, size_t ws_size,
                              hipStream_t stream) {
  const float* x       = (const float*)d_in[0];
  const float* w1_w    = (const float*)d_in[1];
  const float* w1_b    = (const float*)d_in[2];
  const float* w2_w    = (const float*)d_in[3];
  const float* w2_b    = (const float*)d_in[4];
  const float* model_w = (const float*)d_in[5];
  const float* model_b = (const float*)d_in[6];
  float* out = (float*)d_out;

  float* v1 = (float*)d_ws;                 // [B, D]
  float* v2 = v1 + (size_t)B_DIM * D_DIM;   // [B, DOUT]

  // Stage 1: gating vectors (65536 outputs, 256 threads/block)
  gate_kernel<<<(B_DIM * (D_DIM + DOUT)) / 256, 256, 0, stream>>>(
      x, w1_w, w1_b, w2_w, w2_b, v1, v2);

  // Stage 2: fused gated batched GEMM
  dim3 grid(DOUT / TN, S_DIM / TM, B_DIM);  // (32, 4, 32) = 4096 workgroups
  gated_gemm_kernel<<<grid, 256, 0, stream>>>(x, v1, v2, model_w, model_b, out);
}